// Fusion_Conv_att_11149735100590
// MI455X (gfx1250) — compile-verified
//
#include <hip/hip_runtime.h>
#include <hip/hip_bf16.h>

typedef __attribute__((ext_vector_type(2))) float v2f;
typedef __attribute__((ext_vector_type(8))) float v8f;

#define TILE_M 128
#define TILE_N 128
#define TILE_K 16
#define LDA_S  20     // A LDS row stride (floats): 16 + 4 pad, conflict-free for b64 frag reads
#define LDBT   288    // B LDS pair-row stride (floats): 2*128 + 32; 288 % 64 == 32 -> half-waves
                      // hit disjoint bank halves on b64 frag reads

// ---- CDNA5 async global->LDS copy (ASYNCcnt-tracked, no VGPR round trip) ----
__device__ __forceinline__ void async_copy_b128(void* lds_dst, const void* gsrc)
{
    uint32_t l = (uint32_t)(uintptr_t)lds_dst;   // low 32 bits of generic ptr = LDS offset
    asm volatile("global_load_async_to_lds_b128 %0, %1, off"
                 :: "v"(l), "v"(gsrc) : "memory");
}
__device__ __forceinline__ void wait_async0()
{
    asm volatile("s_wait_asynccnt 0x0" ::: "memory");
}

// Issue async fills for one 128x16 A tile (512 16B chunks, 2 per thread).
__device__ __forceinline__ void issue_a_tile(float* dstLds, const float* Wb,
                                             int bm, int K, int kb, int tid)
{
    #pragma unroll
    for (int i = 0; i < 2; ++i) {
        int idx = tid + i * 256;       // 0..511
        int r   = idx >> 2;            // row 0..127
        int c4  = (idx & 3) << 2;      // col 0,4,8,12
        async_copy_b128(&dstLds[r * LDA_S + c4],
                        &Wb[(size_t)(bm + r) * K + kb + c4]);
    }
}

// Prefetch one 16x128 B tile into registers (2 float4 per thread).
__device__ __forceinline__ void load_b_regs(const float* X, int N, int bn, int kb,
                                            int tid, float4& b0, float4& b1)
{
    int i0 = tid, i1 = tid + 256;
    b0 = *(const float4*)&X[(size_t)(kb + (i0 >> 5)) * N + bn + ((i0 & 31) << 2)];
    b1 = *(const float4*)&X[(size_t)(kb + (i1 >> 5)) * N + bn + ((i1 & 31) << 2)];
}

// Store B tile K-pair-interleaved: Bs[k/2][2*n + (k&1)] -> lane K-pair is one b64.
__device__ __forceinline__ void store_b_tile(float* dstLds, int tid,
                                             const float4& b0, const float4& b1)
{
    #pragma unroll
    for (int i = 0; i < 2; ++i) {
        int idx = tid + i * 256;
        int r   = idx >> 5;            // row 0..15
        int c4  = (idx & 31) << 2;     // col 0..124
        const float4& v = i ? b1 : b0;
        float* base = &dstLds[(r >> 1) * LDBT + (r & 1)];
        base[2 * (c4 + 0)] = v.x;
        base[2 * (c4 + 1)] = v.y;
        base[2 * (c4 + 2)] = v.z;
        base[2 * (c4 + 3)] = v.w;
    }
}

// Y[b] = W[b] (MxK, row-major) * X (KxN, row-major) + bias[b]
// Block 128x128x16; 256 threads = 8 waves (4 M-strips x 2 N-strips), wave = 32x64
// via 2x4 grid of 16x16 f32 WMMA accumulators. Double-buffered LDS; A tile comes
// in over global_load_async_to_lds_b128, B tile is register-prefetched + interleaved.
__global__ __launch_bounds__(256)
void gemm_wmma_f32(const float* __restrict__ W, const float* __restrict__ X,
                   const float* __restrict__ bias, float* __restrict__ Y,
                   int M, int N, int K)
{
    __shared__ float As[2][TILE_M * LDA_S];           // 2 x 10240 B
    __shared__ float Bs[2][(TILE_K / 2) * LDBT];      // 2 x  9216 B

    const int bat = blockIdx.z;
    const float* Wb = W + (size_t)bat * M * K;
    const float* bb = bias + (size_t)bat * M;
    float* Yb = Y + (size_t)bat * M * N;

    const int bm = blockIdx.y * TILE_M;
    const int bn = blockIdx.x * TILE_N;

    const int tid  = threadIdx.x;
    const int lane = tid & 31;
    const int wave = tid >> 5;     // 0..7
    const int wm   = wave >> 1;    // 0..3 : 32-row strip
    const int wn   = wave & 1;     // 0..1 : 64-col strip
    const int hh   = lane >> 4;    // half-wave: K pair select / M+8 rows
    const int l16  = lane & 15;

    v8f acc[2][4];
    #pragma unroll
    for (int i = 0; i < 2; ++i)
        #pragma unroll
        for (int j = 0; j < 4; ++j)
            acc[i][j] = (v8f){0.f,0.f,0.f,0.f,0.f,0.f,0.f,0.f};

    // ---- prologue: fill buffer 0 ----
    float4 br0, br1;
    issue_a_tile(As[0], Wb, bm, K, 0, tid);
    load_b_regs(X, N, bn, 0, tid, br0, br1);
    store_b_tile(Bs[0], tid, br0, br1);
    wait_async0();
    __syncthreads();

    const int NT = K / TILE_K;
    for (int t = 0; t < NT; ++t) {
        const int cur = t & 1;
        const int nxt = cur ^ 1;
        const bool hasNext = (t + 1) < NT;
        if (hasNext) {
            issue_a_tile(As[nxt], Wb, bm, K, (t + 1) * TILE_K, tid);
            load_b_regs(X, N, bn, (t + 1) * TILE_K, tid, br0, br1);
        }

        const float* Asb = As[cur];
        const float* Bsb = Bs[cur];
        #pragma unroll
        for (int kk = 0; kk < TILE_K; kk += 4) {
            v2f afr[2];
            #pragma unroll
            for (int i = 0; i < 2; ++i)
                afr[i] = *(const v2f*)&Asb[(wm * 32 + i * 16 + l16) * LDA_S + kk + 2 * hh];
            v2f bfr[4];
            #pragma unroll
            for (int j = 0; j < 4; ++j)
                bfr[j] = *(const v2f*)&Bsb[((kk >> 1) + hh) * LDBT
                                           + 2 * (wn * 64 + j * 16 + l16)];
            #pragma unroll
            for (int i = 0; i < 2; ++i)
                #pragma unroll
                for (int j = 0; j < 4; ++j)
                    acc[i][j] = __builtin_amdgcn_wmma_f32_16x16x4_f32(
                        false, afr[i], false, bfr[j], (short)0, acc[i][j], false, false);
        }

        if (hasNext) store_b_tile(Bs[nxt], tid, br0, br1);
        wait_async0();
        __syncthreads();
    }

    // ---- epilogue: D VGPR v holds row (v + 8*hh), lane l16 holds col ----
    #pragma unroll
    for (int i = 0; i < 2; ++i) {
        #pragma unroll
        for (int j = 0; j < 4; ++j) {
            #pragma unroll
            for (int v = 0; v < 8; ++v) {
                int m = bm + wm * 32 + i * 16 + v + 8 * hh;
                int n = bn + wn * 64 + j * 16 + l16;
                Yb[(size_t)m * N + n] = acc[i][j][v] + bb[m];
            }
        }
    }
}

#define HN 4194304ULL   // 1024*4096

// Partial column-dot: 64 rows of c per block-row, 4 heads at once.
__global__ __launch_bounds__(256)
void attn_score_kernel(const float* __restrict__ Lq, const float* __restrict__ keys,
                       float* __restrict__ partial)
{
    int n  = blockIdx.x * 256 + threadIdx.x;   // 0..4095
    int ch = blockIdx.y;                        // 0..15
    float a0 = 0.f, a1 = 0.f, a2 = 0.f, a3 = 0.f;
    int c0 = ch * 64;
    for (int c = c0; c < c0 + 64; ++c) {
        size_t base = (size_t)c * 4096 + n;
        float q = Lq[base];
        a0 += keys[base] * q;
        a1 += keys[HN + base] * q;
        a2 += keys[2 * HN + base] * q;
        a3 += keys[3 * HN + base] * q;
    }
    size_t pb = ((size_t)ch * 4) * 4096 + n;
    partial[pb] = a0; partial[pb + 4096] = a1;
    partial[pb + 8192] = a2; partial[pb + 12288] = a3;
}

__global__ __launch_bounds__(256)
void attn_softmax_kernel(const float* __restrict__ partial, float* __restrict__ weight,
                         float* __restrict__ wmap)
{
    int n = blockIdx.x * 256 + threadIdx.x;
    float s0 = 0.f, s1 = 0.f, s2 = 0.f, s3 = 0.f;
    #pragma unroll
    for (int ch = 0; ch < 16; ++ch) {
        size_t pb = ((size_t)ch * 4) * 4096 + n;
        s0 += partial[pb]; s1 += partial[pb + 4096];
        s2 += partial[pb + 8192]; s3 += partial[pb + 12288];
    }
    const float inv = 0.03125f;   // 1/sqrt(1024)
    s0 *= inv; s1 *= inv; s2 *= inv; s3 *= inv;
    float m = fmaxf(fmaxf(s0, s1), fmaxf(s2, s3));
    float e0 = __expf(s0 - m), e1 = __expf(s1 - m), e2 = __expf(s2 - m), e3 = __expf(s3 - m);
    float r = 1.f / (e0 + e1 + e2 + e3);
    float w0 = e0 * r, w1 = e1 * r, w2 = e2 * r, w3 = e3 * r;
    weight[n] = w0; weight[4096 + n] = w1; weight[8192 + n] = w2; weight[12288 + n] = w3;
    wmap[n] = w0; wmap[4096 + n] = w1; wmap[8192 + n] = w2; wmap[12288 + n] = w3;
}

// s = L_query + sum_h w[h,n] * keys[h,c,n]   (in-place into L_query buffer)
__global__ __launch_bounds__(256)
void attn_z_kernel(float* __restrict__ S, const float* __restrict__ keys,
                   const float* __restrict__ weight)
{
    size_t idx = (size_t)blockIdx.x * 256 + threadIdx.x;  // < 1024*4096
    int n = (int)(idx & 4095);
    float z = weight[n]            * keys[idx]
            + weight[4096 + n]     * keys[HN + idx]
            + weight[8192 + n]     * keys[2 * HN + idx]
            + weight[12288 + n]    * keys[3 * HN + idx];
    S[idx] += z;
}

// LayerNorm over N=4096 per row; one 256-thread block per row.
__global__ __launch_bounds__(256)
void layernorm_kernel(const float* __restrict__ S, const float* __restrict__ g,
                      const float* __restrict__ bta, float* __restrict__ out)
{
    __shared__ float rsum[256];
    __shared__ float rsq[256];
    const int tid = threadIdx.x;
    const size_t row = (size_t)blockIdx.x * 4096;
    float x[16];
    float sum = 0.f, sq = 0.f;
    #pragma unroll
    for (int i = 0; i < 16; ++i) {
        x[i] = S[row + tid + i * 256];
        sum += x[i]; sq += x[i] * x[i];
    }
    rsum[tid] = sum; rsq[tid] = sq;
    __syncthreads();
    for (int s = 128; s > 0; s >>= 1) {
        if (tid < s) { rsum[tid] += rsum[tid + s]; rsq[tid] += rsq[tid + s]; }
        __syncthreads();
    }
    const float mu  = rsum[0] * (1.f / 4096.f);
    const float var = rsq[0] * (1.f / 4096.f) - mu * mu;
    const float rs  = rsqrtf(var + 1e-5f);
    #pragma unroll
    for (int i = 0; i < 16; ++i) {
        int n = tid + i * 256;
        out[row + n] = (x[i] - mu) * rs * g[n] + bta[n];
    }
}

extern "C" void kernel_launch(void* const* d_in, const int* in_sizes, int n_in,
                              void* d_out, int out_size, void* d_ws, size_t ws_size,
                              hipStream_t stream)
{
    const float* P     = (const float*)d_in[0];   // point_features [1,1024,4096]
    const float* I     = (const float*)d_in[1];   // img_features   [1,1024,4096]
    const float* W_img = (const float*)d_in[2];   // [4,1024,1024]
    const float* b_img = (const float*)d_in[3];   // [4,1024]
    const float* W_L   = (const float*)d_in[4];   // [1024,1024]
    const float* b_L   = (const float*)d_in[5];   // [1024]
    const float* ln_g  = (const float*)d_in[6];   // [4096]
    const float* ln_b  = (const float*)d_in[7];   // [4096]
    const float* W_f   = (const float*)d_in[8];   // [1024,1024]
    const float* b_f   = (const float*)d_in[9];   // [1024]

    float* out  = (float*)d_out;                  // [1024*4096] fusion + [4*4096] weightmap
    float* ws   = (float*)d_ws;

    float* Lq      = ws;                          // 4,194,304 floats (reused as s = z+Lq)
    float* keys    = ws + 4194304;                // 16,777,216 floats (head 0 reused as fused)
    float* partial = ws + 20971520;               // 16*4*4096 floats
    float* weight  = partial + 262144;            // 4*4096 floats
    float* fused   = keys;
    float* wmap    = out + 1024 * 4096;

    dim3 blk(256);

    // 1) L_query = W_L * P + b_L
    gemm_wmma_f32<<<dim3(32, 8, 1), blk, 0, stream>>>(W_L, P, b_L, Lq, 1024, 4096, 1024);
    // 2) keys[h] = W_img[h] * I + b_img[h], 4 heads batched on grid.z
    gemm_wmma_f32<<<dim3(32, 8, 4), blk, 0, stream>>>(W_img, I, b_img, keys, 1024, 4096, 1024);
    // 3) scores -> softmax -> z (+L_query, in-place), weightmap out
    attn_score_kernel<<<dim3(16, 16), blk, 0, stream>>>(Lq, keys, partial);
    attn_softmax_kernel<<<16, blk, 0, stream>>>(partial, weight, wmap);
    attn_z_kernel<<<16384, blk, 0, stream>>>(Lq, keys, weight);
    // 4) LayerNorm over N
    layernorm_kernel<<<1024, blk, 0, stream>>>(Lq, ln_g, ln_b, fused);
    // 5) fusion = W_f * fused + b_f
    gemm_wmma_f32<<<dim3(32, 8, 1), blk, 0, stream>>>(W_f, fused, b_f, out, 1024, 4096, 1024);
}